// CustomLoss_90400471646351
// MI455X (gfx1250) — compile-verified
//
#include <hip/hip_runtime.h>

typedef __attribute__((ext_vector_type(2))) float v2f;
typedef __attribute__((ext_vector_type(8))) float v8f;

#define TILE 16
#define COL_CHUNKS 8
#define WAVES_PER_BLOCK 8
#define INF_BITS 0x7f800000u

// Raw min ops: avoid llvm.minnum's canonicalize insertion (inputs are finite).
__device__ __forceinline__ float vmin(float a, float b) {
    float r;
    asm("v_min_num_f32 %0, %1, %2" : "=v"(r) : "v"(a), "v"(b));
    return r;
}
__device__ __forceinline__ float vmin3(float a, float b, float c) {
    float r;
    asm("v_min3_num_f32 %0, %1, %2, %3" : "=v"(r) : "v"(a), "v"(b), "v"(c));
    return r;
}
// max(min(a,b), 0) in one VOP3: v_minmax_num_f32 dst, a, b, 0
__device__ __forceinline__ float vminmax0(float a, float b) {
    float r;
    asm("v_minmax_num_f32 %0, %1, %2, 0" : "=v"(r) : "v"(a), "v"(b));
    return r;
}
__device__ __forceinline__ float vmax0(float a) {
    float r;
    asm("v_max_num_f32 %0, %1, 0" : "=v"(r) : "v"(a));
    return r;
}

// ---------------------------------------------------------------------------
// Kernel 1: build augmented operands and reset min arrays.
//   Apad[i] = [-2*px, -2*py, px^2+py^2, 1]   (A rows = p points)
//   Bpad[j] = [tx, ty, 1, tx^2+ty^2]         (B cols = t points)
// (A x B)[i][j] = ||p_i||^2 + ||t_j||^2 - 2 p_i.t_j == squared distance.
// ---------------------------------------------------------------------------
__global__ void prep_kernel(const float* __restrict__ yp, const float* __restrict__ yt,
                            float* __restrict__ Apad, float* __restrict__ Bpad,
                            unsigned* __restrict__ rowmin, unsigned* __restrict__ colmin,
                            int N, int M) {
    int i = blockIdx.x * blockDim.x + threadIdx.x;
    if (i < N) {
        float px = yp[2 * i], py = yp[2 * i + 1];
        Apad[4 * i + 0] = -2.0f * px;
        Apad[4 * i + 1] = -2.0f * py;
        Apad[4 * i + 2] = px * px + py * py;
        Apad[4 * i + 3] = 1.0f;
        rowmin[i] = INF_BITS;
    }
    if (i < M) {
        float tx = yt[2 * i], ty = yt[2 * i + 1];
        Bpad[4 * i + 0] = tx;
        Bpad[4 * i + 1] = ty;
        Bpad[4 * i + 2] = 1.0f;
        Bpad[4 * i + 3] = tx * tx + ty * ty;
        colmin[i] = INF_BITS;
    }
}

// ---------------------------------------------------------------------------
// Kernel 2: WMMA sweep over SQUARED distances (sqrt deferred: monotonic).
// One wave = TWO 16-row tiles (32 p-rows) x (M/COL_CHUNKS) t-cols: each B
// fragment load feeds two wmmas (A0, A1), halving loads and column atomics.
// A 16x4 f32 layout: lane (h*16+l) holds A[l][2h], A[l][2h+1]
// B 4x16  f32 layout: lane (h*16+l) holds B[2h][l], B[2h+1][l]
// C/D 16x16 f32:      VGPR v, lane (h*16+l) -> D[v+8h][l]
// ---------------------------------------------------------------------------
template <int TPC_CT>   // compile-time tiles-per-chunk (0 => runtime)
__global__ __launch_bounds__(256) void chamfer_wmma_kernel(
    const float* __restrict__ Apad, const float* __restrict__ Bpad,
    unsigned* __restrict__ rowmin, unsigned* __restrict__ colmin,
    int colsPerChunk) {
    const int lane = threadIdx.x & 31;
    const int lid  = lane & 15;
    const int half = lane >> 4;

    const int wid   = blockIdx.x * WAVES_PER_BLOCK + (threadIdx.x >> 5);
    const int rpair = wid / COL_CHUNKS;          // 32-row pair index
    const int chunk = wid % COL_CHUNKS;
    const int row0  = rpair * (2 * TILE);
    const int cbase = chunk * colsPerChunk;
    const int tpc   = TPC_CT ? TPC_CT : (colsPerChunk / TILE);

    // Two A fragments, invariant for the wave's whole strip.
    v2f a0 = *(const v2f*)(Apad + (size_t)(row0 + lid) * 4 + 2 * half);
    v2f a1 = *(const v2f*)(Apad + (size_t)(row0 + TILE + lid) * 4 + 2 * half);

    // Per-lane B pointer; each tile advances 16 points * 4 floats = 64 floats.
    const float* bptr = Bpad + (size_t)(cbase + lid) * 4 + 2 * half;

    v8f rmin0, rmin1;
#pragma unroll
    for (int v = 0; v < 8; ++v) { rmin0[v] = __builtin_inff(); rmin1[v] = __builtin_inff(); }

    // Software pipeline: prefetch next B fragment before consuming current.
    // (One-tile overread at the end lands in rowmin/colmin storage — same
    //  workspace allocation, value unused.)
    v2f bcur = *(const v2f*)bptr;
    for (int t = 0; t < tpc; ++t) {
        v2f bnext = *(const v2f*)(bptr + (size_t)(t + 1) * 64);

        v8f c0 = {}, c1 = {};
        c0 = __builtin_amdgcn_wmma_f32_16x16x4_f32(false, a0, false, bcur,
                                                   (short)0, c0, false, false);
        c1 = __builtin_amdgcn_wmma_f32_16x16x4_f32(false, a1, false, bcur,
                                                   (short)0, c1, false, false);
        bcur = bnext;

        // Row accumulators: 16 raw v_min ops (each folds one value per lane).
#pragma unroll
        for (int v = 0; v < 8; ++v) {
            rmin0[v] = vmin(rmin0[v], c0[v]);
            rmin1[v] = vmin(rmin1[v], c1[v]);
        }
        // Column min of this tile-pair: 16 values -> min3 chain + fused clamp.
        float cmin = vmin3(c0[0], c1[0], c0[1]);
        cmin = vmin3(cmin, c1[1], c0[2]);
        cmin = vmin3(cmin, c1[2], c0[3]);
        cmin = vmin3(cmin, c1[3], c0[4]);
        cmin = vmin3(cmin, c1[4], c0[5]);
        cmin = vmin3(cmin, c1[5], c0[6]);
        cmin = vmin3(cmin, c1[6], c0[7]);
        cmin = vminmax0(cmin, c1[7]);   // max(min(...), 0)

        // Both halves hold the same column (different row sets): both fold in
        // with a no-return atomic — no wait, no cross-lane ds op in the loop.
        // cmin >= 0, so uint bit ordering == float ordering.
        atomicMin(&colmin[cbase + t * TILE + lid], __float_as_uint(cmin));
    }

    // Row mins: butterfly across the 16 lanes of each half-wave (once per wave).
#pragma unroll
    for (int v = 0; v < 8; ++v) {
        float r0 = rmin0[v], r1 = rmin1[v];
        r0 = vmin(r0, __shfl_xor(r0, 1, 32));
        r1 = vmin(r1, __shfl_xor(r1, 1, 32));
        r0 = vmin(r0, __shfl_xor(r0, 2, 32));
        r1 = vmin(r1, __shfl_xor(r1, 2, 32));
        r0 = vmin(r0, __shfl_xor(r0, 4, 32));
        r1 = vmin(r1, __shfl_xor(r1, 4, 32));
        r0 = vmin(r0, __shfl_xor(r0, 8, 32));
        r1 = vmin(r1, __shfl_xor(r1, 8, 32));
        if (lid == 0) {
            atomicMin(&rowmin[row0 + 8 * half + v], __float_as_uint(vmax0(r0)));
            atomicMin(&rowmin[row0 + TILE + 8 * half + v], __float_as_uint(vmax0(r1)));
        }
    }
}

// ---------------------------------------------------------------------------
// Kernel 3: sum(sqrt(rowmin)) + sum(sqrt(colmin)), scale by 1/M.
// ---------------------------------------------------------------------------
__global__ __launch_bounds__(256) void final_reduce_kernel(
    const unsigned* __restrict__ rowmin, const unsigned* __restrict__ colmin,
    float* __restrict__ out, int N, int M) {
    __shared__ float sdata[256];
    float s = 0.0f;
    for (int i = threadIdx.x; i < N; i += 256)
        s += __builtin_amdgcn_sqrtf(__uint_as_float(rowmin[i]));
    for (int i = threadIdx.x; i < M; i += 256)
        s += __builtin_amdgcn_sqrtf(__uint_as_float(colmin[i]));
    sdata[threadIdx.x] = s;
    __syncthreads();
    for (int off = 128; off > 0; off >>= 1) {
        if ((int)threadIdx.x < off) sdata[threadIdx.x] += sdata[threadIdx.x + off];
        __syncthreads();
    }
    if (threadIdx.x == 0) out[0] = sdata[0] / (float)M;
}

extern "C" void kernel_launch(void* const* d_in, const int* in_sizes, int n_in,
                              void* d_out, int out_size, void* d_ws, size_t ws_size,
                              hipStream_t stream) {
    const float* yp = (const float*)d_in[0];
    const float* yt = (const float*)d_in[1];
    float* out = (float*)d_out;

    const int N = in_sizes[0] / 2;   // 8192 p points
    const int M = in_sizes[1] / 2;   // 8192 t points

    // Workspace layout: Apad | Bpad | rowmin | colmin
    char* ws = (char*)d_ws;
    float*    Apad   = (float*)ws;                               // N*4 floats
    float*    Bpad   = (float*)(ws + (size_t)N * 16);            // M*4 floats
    unsigned* rowmin = (unsigned*)(ws + (size_t)N * 16 + (size_t)M * 16);
    unsigned* colmin = rowmin + N;

    const int mx = (N > M) ? N : M;
    prep_kernel<<<(mx + 255) / 256, 256, 0, stream>>>(yp, yt, Apad, Bpad,
                                                      rowmin, colmin, N, M);

    const int rowPairs     = N / (2 * TILE);               // 256
    const int colsPerChunk = M / COL_CHUNKS;               // 1024
    const int totalWaves   = rowPairs * COL_CHUNKS;        // 2048
    const int blocks       = totalWaves / WAVES_PER_BLOCK; // 256

    if (colsPerChunk == 64 * TILE) {
        chamfer_wmma_kernel<64><<<blocks, 256, 0, stream>>>(Apad, Bpad, rowmin,
                                                            colmin, colsPerChunk);
    } else {
        chamfer_wmma_kernel<0><<<blocks, 256, 0, stream>>>(Apad, Bpad, rowmin,
                                                           colmin, colsPerChunk);
    }

    final_reduce_kernel<<<1, 256, 0, stream>>>(rowmin, colmin, out, N, M);
}